// ReadHead_72524817760564
// MI455X (gfx1250) — compile-verified
//
#include <hip/hip_runtime.h>

typedef __attribute__((ext_vector_type(2))) float v2f;
typedef __attribute__((ext_vector_type(8))) float v8f;

constexpr int Hc   = 128;  // memory cells
constexpr int Dc   = 20;   // cell dim
constexpr int Ic   = 8;    // controller dim
constexpr int ROWS = 16;   // batch rows per block (one WMMA M-tile)
constexpr int NT   = 128;  // threads per block = 4 wave32

// ---- wave32 block reductions: shfl within wave, LDS combine across 4 waves ----
__device__ __forceinline__ float blkRedSum(float v, volatile float* scr, int wave, int lane) {
#pragma unroll
  for (int o = 16; o > 0; o >>= 1) v += __shfl_xor(v, o, 32);
  __syncthreads();
  if (lane == 0) scr[wave] = v;
  __syncthreads();
  return scr[0] + scr[1] + scr[2] + scr[3];
}
__device__ __forceinline__ float blkRedMax(float v, volatile float* scr, int wave, int lane) {
#pragma unroll
  for (int o = 16; o > 0; o >>= 1) v = fmaxf(v, __shfl_xor(v, o, 32));
  __syncthreads();
  if (lane == 0) scr[wave] = v;
  __syncthreads();
  return fmaxf(fmaxf(scr[0], scr[1]), fmaxf(scr[2], scr[3]));
}

// ---- one 16x16 output tile of (16 x K) @ (K x 128) via V_WMMA_F32_16X16X4_F32 ----
// Al : 16 rows x K (row-major, stride K)      A[m,k]
// Bl : 128 cols x K (row-major by column)    B[k,n] stored as Bl[n*K + k]
// Ol : 16 x 128 row-major output; `tile` selects columns [16*tile, 16*tile+16)
// ISA 32-bit A 16x4 layout: lanes 0-15 -> (K=0,K=1) in v0/v1, lanes 16-31 -> (K=2,K=3).
// C/D layout: VGPR r -> row r (lanes 0-15) / row r+8 (lanes 16-31), N = lane&15.
template <int K>
__device__ __forceinline__ void gemm16_tile(const float* Al, const float* Bl, float* Ol,
                                            int tile, int lane) {
  const int half  = lane >> 4;    // 0: K pair {0,1}, 1: K pair {2,3}
  const int l15   = lane & 15;
  const int koff  = half * 2;
  const int cbase = tile * 16;
  v8f acc = {};
#pragma unroll
  for (int k0 = 0; k0 < K; k0 += 4) {        // K=20 -> 5 steps, K=8 -> 2 steps
    v2f a, b;
    a.x = Al[l15 * K + k0 + koff];
    a.y = Al[l15 * K + k0 + koff + 1];
    b.x = Bl[(cbase + l15) * K + k0 + koff];
    b.y = Bl[(cbase + l15) * K + k0 + koff + 1];
    acc = __builtin_amdgcn_wmma_f32_16x16x4_f32(false, a, false, b,
                                                (short)0, acc, false, false);
  }
#pragma unroll
  for (int r = 0; r < 8; ++r)
    Ol[(r + half * 8) * Hc + cbase + l15] = acc[r];
}

__global__ __launch_bounds__(NT)
void ntm_readhead_kernel(const float* __restrict__ mem,    // (B,H,D)
                         const float* __restrict__ prev,   // (B,H)
                         const float* __restrict__ hid,    // (B,I)
                         const float* __restrict__ Wk,  const float* __restrict__ bk,
                         const float* __restrict__ Wbeta, const float* __restrict__ bbeta,
                         const float* __restrict__ Wg,  const float* __restrict__ bg,
                         const float* __restrict__ Ws,  const float* __restrict__ bs,
                         float* __restrict__ out_read,     // (B,D)
                         float* __restrict__ out_wn,       // (B,H)
                         int Btot) {
  __shared__ float ml[Hc * Dc];      // memory[B-1]  (128x20, 10KB)
  __shared__ float keyA[ROWS * Dc];  // relu keys    (16x20)
  __shared__ float hA[ROWS * Ic];    // hidden rows  (16x8)
  __shared__ float WsL[Hc * Ic];     // Ws           (128x8)
  __shared__ float cosM[ROWS * Hc];  // WMMA out 1
  __shared__ float slog[ROWS * Hc];  // WMMA out 2
  __shared__ float bsl[Hc];
  __shared__ float betas[ROWS], gsv[ROWS];
  __shared__ float wgL[Hc], shiftL[Hc], wnL[Hc];
  __shared__ float part[4 * Dc];
  __shared__ float scr[4];

  const int tid  = threadIdx.x;
  const int wave = tid >> 5, lane = tid & 31;
  // wave index as a wave-uniform scalar -> WMMA tile selection is straight-line,
  // no exec-masked vector loop around the matrix ops (EXEC stays all-1s).
  const int wave_u = __builtin_amdgcn_readfirstlane(wave);
  const int b0   = blockIdx.x * ROWS;

  // ---- stage shared operands into LDS ----
  hA[tid]  = hid[(size_t)b0 * Ic + tid];   // 16*8 == 128, coalesced
  bsl[tid] = bs[tid];
  for (int i = tid; i < Hc * Ic; i += NT) WsL[i] = Ws[i];
  const float* mlast = mem + (size_t)(Btot - 1) * Hc * Dc;
  for (int i = tid; i < Hc * Dc; i += NT) ml[i] = mlast[i];
  __syncthreads();

  // per-row scalars: beta and g (== gamma, faithful to reference)
  if (tid < ROWS) {
    float sb = bbeta[0], sg = bg[0];
#pragma unroll
    for (int i = 0; i < Ic; ++i) {
      sb += hA[tid * Ic + i] * Wbeta[i];
      sg += hA[tid * Ic + i] * Wg[i];
    }
    betas[tid] = sb;
    gsv[tid]   = sg;
  }
  // key = relu(h @ Wk.T + bk): 16x20 entries
  for (int idx = tid; idx < ROWS * Dc; idx += NT) {
    int r = idx / Dc, d = idx % Dc;
    float s = bk[d];
#pragma unroll
    for (int i = 0; i < Ic; ++i) s += hA[r * Ic + i] * Wk[d * Ic + i];
    keyA[idx] = fmaxf(s, 0.f);
  }
  __syncthreads();

  // ---- the two shared-weight GEMMs on the matrix core (2 tiles per wave) ----
  gemm16_tile<Dc>(keyA, ml,  cosM, wave_u,     lane);  // cos = key @ memory[B-1].T
  gemm16_tile<Dc>(keyA, ml,  cosM, wave_u + 4, lane);
  gemm16_tile<Ic>(hA,   WsL, slog, wave_u,     lane);  // shift logits = h @ Ws.T
  gemm16_tile<Ic>(hA,   WsL, slog, wave_u + 4, lane);
  __syncthreads();

  // ---- per-row pipeline, thread = memory cell h ----
  for (int r = 0; r < ROWS; ++r) {
    const int   b    = b0 + r;
    const float beta = betas[r];
    const float g    = gsv[r];
    const int   h    = tid;

    // content softmax
    float x  = cosM[r * Hc + h] * beta;
    float m  = blkRedMax(x, scr, wave, lane);
    float e  = __expf(x - m);
    float s  = blkRedSum(e, scr, wave, lane);
    float wc = e / s;
    float wg = g * wc + (1.f - g) * prev[(size_t)b * Hc + h];

    // shift softmax
    float y  = slog[r * Hc + h] + bsl[h];
    float m2 = blkRedMax(y, scr, wave, lane);
    float e2 = __expf(y - m2);
    float s2 = blkRedSum(e2, scr, wave, lane);

    wgL[h]    = wg;
    shiftL[h] = e2 / s2;
    __syncthreads();

    // circular convolution: circ[h] = sum_j shift[(h-j) mod H] * w_g[j]
    float c = 0.f;
#pragma unroll 4
    for (int j = 0; j < Hc; ++j)
      c += shiftL[(h - j) & (Hc - 1)] * wgL[j];   // broadcast + conflict-free

    // sharpen (gamma == g, faithful) and renormalize
    float wp = __powf(c, g);
    float s3 = blkRedSum(wp, scr, wave, lane);
    float wn = wp / s3;
    wnL[h] = wn;
    out_wn[(size_t)b * Hc + h] = wn;
    __syncthreads();

    // read[b,d] = sum_h w_next[h] * memory[b,h,d]
    // wave w covers h in [32w, 32w+32); lane = d (<20). Each wave walks a
    // contiguous 2560B span of the slab -> cachelines fully consumed.
    if (lane < Dc) {
      float acc = 0.f;
      const float* mrow = mem + (size_t)b * Hc * Dc + (wave * 32) * Dc + lane;
#pragma unroll 8
      for (int hh = 0; hh < 32; ++hh)
        acc += wnL[wave * 32 + hh] * mrow[hh * Dc];
      part[wave * Dc + lane] = acc;
    }
    __syncthreads();
    if (tid < Dc)
      out_read[(size_t)b * Dc + tid] =
          part[tid] + part[Dc + tid] + part[2 * Dc + tid] + part[3 * Dc + tid];
    __syncthreads();
  }
}

extern "C" void kernel_launch(void* const* d_in, const int* in_sizes, int n_in,
                              void* d_out, int out_size, void* d_ws, size_t ws_size,
                              hipStream_t stream) {
  const float* mem   = (const float*)d_in[0];
  const float* prev  = (const float*)d_in[1];
  const float* hid   = (const float*)d_in[2];
  const float* Wk    = (const float*)d_in[3];
  const float* bk    = (const float*)d_in[4];
  const float* Wbeta = (const float*)d_in[5];
  const float* bbeta = (const float*)d_in[6];
  const float* Wg    = (const float*)d_in[7];
  const float* bg    = (const float*)d_in[8];
  const float* Ws    = (const float*)d_in[9];
  const float* bs    = (const float*)d_in[10];

  const int Btot = in_sizes[0] / (Hc * Dc);   // 16384
  float* out_read = (float*)d_out;
  float* out_wn   = out_read + (size_t)Btot * Dc;

  const int grid = Btot / ROWS;               // 1024 blocks
  hipLaunchKernelGGL(ntm_readhead_kernel, dim3(grid), dim3(NT), 0, stream,
                     mem, prev, hid, Wk, bk, Wbeta, bbeta, Wg, bg, Ws, bs,
                     out_read, out_wn, Btot);
}